// WassersteinGaussian_34591666602825
// MI455X (gfx1250) — compile-verified
//
#include <hip/hip_runtime.h>
#include <cstdint>

#define W2_EPS 1e-8f
#define BLK 256

// ---------------------------------------------------------------------------
// CDNA5 async global->LDS copy (ASYNCcnt-tracked per-lane 128-bit copy).
// Builtin signature (from hipcc diagnostic): params are v4i pointers,
// addrspace(1) source first, addrspace(3) LDS dest, imm offset, imm cpol.
// ---------------------------------------------------------------------------
#if __has_builtin(__builtin_amdgcn_global_load_async_to_lds_b128)
#define HAS_ASYNC_LDS 1
#else
#define HAS_ASYNC_LDS 0
#endif

#if HAS_ASYNC_LDS
typedef int w2_v4i __attribute__((vector_size(4 * sizeof(int))));
typedef __attribute__((address_space(1))) w2_v4i* w2_gptr;
typedef __attribute__((address_space(3))) w2_v4i* w2_lptr;
#endif

__device__ __forceinline__ void copy16_to_lds(const float* __restrict__ gsrc,
                                              float* __restrict__ ldst) {
#if HAS_ASYNC_LDS
    w2_gptr g = (w2_gptr)(unsigned long long)(uintptr_t)gsrc;
    // Low 32 bits of a generic LDS pointer are the LDS byte offset (aperture
    // mapping truncates to the low 32 bits).
    w2_lptr l = (w2_lptr)(unsigned int)(uintptr_t)ldst;
    __builtin_amdgcn_global_load_async_to_lds_b128(g, l, /*offset=*/0, /*cpol=*/0);
#else
    *(float4*)ldst = *(const float4*)gsrc;
#endif
}

__device__ __forceinline__ void wait_async_then_barrier() {
#if HAS_ASYNC_LDS
#if __has_builtin(__builtin_amdgcn_s_wait_asynccnt)
    __builtin_amdgcn_s_wait_asynccnt(0);
#else
    asm volatile("s_wait_asynccnt 0" ::: "memory");
#endif
#endif
    __syncthreads();
}

// ---------------------------------------------------------------------------
// Per-Gaussian W2 distance: cov algebra + analytic symmetric 3x3 eigenvalues
// (Smith's trigonometric method), all fp32.
// ---------------------------------------------------------------------------
__device__ __forceinline__ float w2_gauss(const float* __restrict__ l1,
                                          const float* __restrict__ s1in,
                                          const float* __restrict__ r1,
                                          const float* __restrict__ l2,
                                          const float* __restrict__ s2in,
                                          const float* __restrict__ r2) {
    const float s1x = fmaxf(s1in[0], W2_EPS);
    const float s1y = fmaxf(s1in[1], W2_EPS);
    const float s1z = fmaxf(s1in[2], W2_EPS);
    const float s2x = fmaxf(s2in[0], W2_EPS);
    const float s2y = fmaxf(s2in[1], W2_EPS);
    const float s2z = fmaxf(s2in[2], W2_EPS);

    const float dx = l1[0] - l2[0];
    const float dy = l1[1] - l2[1];
    const float dz = l1[2] - l2[2];
    const float loc_diff2 = dx * dx + dy * dy + dz * dz;

    // cov2 = R2 diag(s2) R2^T (symmetric, 6 unique entries); R2 row-major.
    const float c00 = s2x * r2[0] * r2[0] + s2y * r2[1] * r2[1] + s2z * r2[2] * r2[2];
    const float c01 = s2x * r2[0] * r2[3] + s2y * r2[1] * r2[4] + s2z * r2[2] * r2[5];
    const float c02 = s2x * r2[0] * r2[6] + s2y * r2[1] * r2[7] + s2z * r2[2] * r2[8];
    const float c11 = s2x * r2[3] * r2[3] + s2y * r2[4] * r2[4] + s2z * r2[5] * r2[5];
    const float c12 = s2x * r2[3] * r2[6] + s2y * r2[4] * r2[7] + s2z * r2[5] * r2[8];
    const float c22 = s2x * r2[6] * r2[6] + s2y * r2[7] * r2[7] + s2z * r2[8] * r2[8];

    // T = cov2 * R1 ; T[j][l] = sum_k cov2[j][k] * R1[k][l]
    float T[3][3];
#pragma unroll
    for (int l = 0; l < 3; ++l) {
        T[0][l] = c00 * r1[l] + c01 * r1[3 + l] + c02 * r1[6 + l];
        T[1][l] = c01 * r1[l] + c11 * r1[3 + l] + c12 * r1[6 + l];
        T[2][l] = c02 * r1[l] + c12 * r1[3 + l] + c22 * r1[6 + l];
    }
    // M = R1^T * T ; M[i][l] = sum_j R1[j][i] * T[j][l]
    float M[3][3];
#pragma unroll
    for (int i = 0; i < 3; ++i)
#pragma unroll
        for (int l = 0; l < 3; ++l)
            M[i][l] = r1[i] * T[0][l] + r1[3 + i] * T[1][l] + r1[6 + i] * T[2][l];

    const float m01 = 0.5f * (M[0][1] + M[1][0]);
    const float m02 = 0.5f * (M[0][2] + M[2][0]);
    const float m12 = 0.5f * (M[1][2] + M[2][1]);

    const float q1x = sqrtf(s1x), q1y = sqrtf(s1y), q1z = sqrtf(s1z);
    // E = diag(sqrt(s1)) M diag(sqrt(s1)), symmetrized, + EPS*I
    const float a = q1x * q1x * M[0][0] + W2_EPS;
    const float b = q1y * q1y * M[1][1] + W2_EPS;
    const float c = q1z * q1z * M[2][2] + W2_EPS;
    const float d = q1x * q1y * m01;
    const float e = q1x * q1z * m02;
    const float f = q1y * q1z * m12;

    // Analytic eigenvalues of symmetric [[a,d,e],[d,b,f],[e,f,c]]
    const float q  = (a + b + c) * (1.0f / 3.0f);
    const float p1 = d * d + e * e + f * f;
    const float aa = a - q, bb = b - q, cc = c - q;
    const float p2 = aa * aa + bb * bb + cc * cc + 2.0f * p1;

    float e0, e1, e2;
    if (p2 <= 1e-30f) {
        e0 = e1 = e2 = q;
    } else {
        const float p     = sqrtf(p2 * (1.0f / 6.0f));
        const float inv_p = 1.0f / p;
        const float b00 = aa * inv_p, b11 = bb * inv_p, b22 = cc * inv_p;
        const float b01 = d * inv_p,  b02 = e * inv_p,  b12 = f * inv_p;
        float r = 0.5f * (b00 * (b11 * b22 - b12 * b12)
                        - b01 * (b01 * b22 - b12 * b02)
                        + b02 * (b01 * b12 - b11 * b02));
        r = fminf(1.0f, fmaxf(-1.0f, r));
        const float phi = acosf(r) * (1.0f / 3.0f);
        e0 = q + 2.0f * p * __cosf(phi);
        e2 = q + 2.0f * p * __cosf(phi + 2.0943951023931953f);  // +2*pi/3
        e1 = 3.0f * q - e0 - e2;
    }
    e0 = fmaxf(e0, W2_EPS);
    e1 = fmaxf(e1, W2_EPS);
    e2 = fmaxf(e2, W2_EPS);
    const float e_sqrt_trace = sqrtf(e0) + sqrtf(e1) + sqrtf(e2);

    float cov_w = (s1x + s1y + s1z) + (c00 + c11 + c22) - 2.0f * e_sqrt_trace;
    cov_w = fmaxf(cov_w, 0.0f);
    return sqrtf(fmaxf(loc_diff2 + cov_w, W2_EPS));
}

// LDS layout (floats): loc1[768] s1[768] rot1[2304] loc2[768] s2[768] rot2[2304]
#define L_LOC1 0
#define L_SC1  768
#define L_ROT1 1536
#define L_LOC2 3840
#define L_SC2  4608
#define L_ROT2 5376
#define L_TOT  7680  // 30720 bytes

__global__ void __launch_bounds__(BLK)
WassersteinGaussian_34591666602825_kernel(const float* __restrict__ loc1,
                                          const float* __restrict__ scale1,
                                          const float* __restrict__ rot1,
                                          const float* __restrict__ loc2,
                                          const float* __restrict__ scale2,
                                          const float* __restrict__ rot2,
                                          float* __restrict__ out, int n) {
    __shared__ float smem[L_TOT];
    const int tid = threadIdx.x;
    const long long g0 = (long long)blockIdx.x * BLK;

    if (g0 + BLK <= (long long)n) {
        // Full block: stage 30 KB via async 128-bit coalesced copies.
        const float* gl1 = loc1 + 3 * g0;
        const float* gs1 = scale1 + 3 * g0;
        const float* gr1 = rot1 + 9 * g0;
        const float* gl2 = loc2 + 3 * g0;
        const float* gs2 = scale2 + 3 * g0;
        const float* gr2 = rot2 + 9 * g0;
#pragma unroll 1
        for (int i = tid; i < 192; i += BLK) copy16_to_lds(gl1 + 4 * i, smem + L_LOC1 + 4 * i);
#pragma unroll 1
        for (int i = tid; i < 192; i += BLK) copy16_to_lds(gs1 + 4 * i, smem + L_SC1 + 4 * i);
#pragma unroll 1
        for (int i = tid; i < 576; i += BLK) copy16_to_lds(gr1 + 4 * i, smem + L_ROT1 + 4 * i);
#pragma unroll 1
        for (int i = tid; i < 192; i += BLK) copy16_to_lds(gl2 + 4 * i, smem + L_LOC2 + 4 * i);
#pragma unroll 1
        for (int i = tid; i < 192; i += BLK) copy16_to_lds(gs2 + 4 * i, smem + L_SC2 + 4 * i);
#pragma unroll 1
        for (int i = tid; i < 576; i += BLK) copy16_to_lds(gr2 + 4 * i, smem + L_ROT2 + 4 * i);
        wait_async_then_barrier();

        out[g0 + tid] = w2_gauss(smem + L_LOC1 + 3 * tid, smem + L_SC1 + 3 * tid,
                                 smem + L_ROT1 + 9 * tid, smem + L_LOC2 + 3 * tid,
                                 smem + L_SC2 + 3 * tid, smem + L_ROT2 + 9 * tid);
    } else {
        // Tail block: direct scalar loads (block-uniform branch; no barrier needed).
        const long long g = g0 + tid;
        if (g < (long long)n) {
            float l1[3], s1[3], r1[9], l2[3], s2[3], r2[9];
#pragma unroll
            for (int k = 0; k < 3; ++k) {
                l1[k] = loc1[3 * g + k];
                s1[k] = scale1[3 * g + k];
                l2[k] = loc2[3 * g + k];
                s2[k] = scale2[3 * g + k];
            }
#pragma unroll
            for (int k = 0; k < 9; ++k) {
                r1[k] = rot1[9 * g + k];
                r2[k] = rot2[9 * g + k];
            }
            out[g] = w2_gauss(l1, s1, r1, l2, s2, r2);
        }
    }
}

extern "C" void kernel_launch(void* const* d_in, const int* in_sizes, int n_in,
                              void* d_out, int out_size, void* d_ws, size_t ws_size,
                              hipStream_t stream) {
    const float* loc1   = (const float*)d_in[0];
    const float* scale1 = (const float*)d_in[1];
    const float* rot1   = (const float*)d_in[2];
    const float* loc2   = (const float*)d_in[3];
    const float* scale2 = (const float*)d_in[4];
    const float* rot2   = (const float*)d_in[5];
    float* out = (float*)d_out;

    const int n = in_sizes[0] / 3;  // loc1 is [B,3]
    const int grid = (n + BLK - 1) / BLK;
    WassersteinGaussian_34591666602825_kernel<<<grid, BLK, 0, stream>>>(
        loc1, scale1, rot1, loc2, scale2, rot2, out, n);
}